// SGC_11441792877213
// MI455X (gfx1250) — compile-verified
//
#include <hip/hip_runtime.h>
#include <hip/hip_bf16.h>

#define IN_DIM 128
#define OUT_DIM 64

typedef __attribute__((ext_vector_type(2))) float v2f;
typedef __attribute__((ext_vector_type(8))) float v8f;

// ---------------- utility: grid-stride float4 zero ----------------
__global__ void sgc_zero_f4(float* __restrict__ p, long n4) {
    long i = (long)blockIdx.x * blockDim.x + threadIdx.x;
    long stride = (long)gridDim.x * blockDim.x;
    float4 z = make_float4(0.f, 0.f, 0.f, 0.f);
    for (; i < n4; i += stride) ((float4*)p)[i] = z;
}

// ---------------- in-degree count (into deg buffer) ----------------
__global__ void sgc_degree(const int* __restrict__ dst, float* __restrict__ deg, int E) {
    int e = blockIdx.x * blockDim.x + threadIdx.x;
    if (e < E) atomicAdd(deg + dst[e], 1.0f);
}

// ---------------- deg -> norm = deg>0 ? rsqrt(deg) : 0 (in place) ----------------
__global__ void sgc_norm(float* __restrict__ deg, int N) {
    int n = blockIdx.x * blockDim.x + threadIdx.x;
    if (n < N) {
        float d = deg[n];
        deg[n] = (d > 0.f) ? rsqrtf(d) : 0.f;
    }
}

// ------- edge scatter: hout[dst] += hin[src] * norm[src]; one wave32 per edge, float4/lane -------
__global__ void __launch_bounds__(256)
sgc_scatter(const int* __restrict__ src, const int* __restrict__ dst,
            const float* __restrict__ hin, const float* __restrict__ norm,
            float* __restrict__ hout, int E) {
    int gt = blockIdx.x * blockDim.x + threadIdx.x;
    int e = gt >> 5;          // wave per edge
    int lane = gt & 31;       // 32 lanes x float4 = 128 floats
    if (e >= E) return;
    int s = src[e];
    int d = dst[e];
    float w = norm[s];
    const float4* row = (const float4*)(hin + (size_t)s * IN_DIM);
    float4 v = row[lane];
    float* orow = hout + (size_t)d * IN_DIM + lane * 4;
    atomicAdd(orow + 0, v.x * w);
    atomicAdd(orow + 1, v.y * w);
    atomicAdd(orow + 2, v.z * w);
    atomicAdd(orow + 3, v.w * w);
}

// ---------------- post-normalize: h[n,:] *= norm[n] ----------------
__global__ void __launch_bounds__(256)
sgc_scale(float* __restrict__ h, const float* __restrict__ norm, int N) {
    int gt = blockIdx.x * blockDim.x + threadIdx.x;
    int n = gt >> 5;
    int lane = gt & 31;
    if (n >= N) return;
    float w = norm[n];
    float4* row = (float4*)(h + (size_t)n * IN_DIM);
    float4 v = row[lane];
    v.x *= w; v.y *= w; v.z *= w; v.w *= w;
    row[lane] = v;
}

// ---------------- final linear: out[N,64] = H[N,128] @ W^T + bias, fp32 WMMA 16x16x4 ----------------
// Fragment layout (CDNA5 ISA 7.12.2, 32-bit):
//   A 16x4: lane<16 -> row M=lane, regs hold K={k,k+1}; lane>=16 -> K={k+2,k+3}
//   B 4x16: lane<16 -> col N=lane, regs hold K={k,k+1}; lane>=16 -> K={k+2,k+3}
//   C/D 16x16: reg r -> (M=r, N=lane) for lane<16, (M=r+8, N=lane-16) for lane>=16
__global__ void __launch_bounds__(256)
sgc_linear_wmma(const float* __restrict__ H, const float* __restrict__ W,
                const float* __restrict__ bias, float* __restrict__ out, int N) {
    __shared__ float Wl[OUT_DIM * IN_DIM];   // 32 KB: weight[out][in]
    __shared__ float Bl[OUT_DIM];

    // cooperative stage of weight + bias into LDS
    for (int i = threadIdx.x; i < (OUT_DIM * IN_DIM) / 4; i += 256)
        ((float4*)Wl)[i] = ((const float4*)W)[i];
    if (threadIdx.x < OUT_DIM) Bl[threadIdx.x] = bias[threadIdx.x];
    __syncthreads();

    int wave = threadIdx.x >> 5;
    int lane = threadIdx.x & 31;
    int tile = blockIdx.x * 8 + wave;                  // 8 waves / block
    int mt = tile >> 2;                                // 4 N-tiles (OUT_DIM/16)
    int nt = tile & 3;
    int m0 = mt * 16, n0 = nt * 16;
    if (m0 >= N) return;

    int r  = lane & 15;
    int kb = (lane >> 4) << 1;                         // 0 or 2
    int arow = m0 + r;
    if (arow >= N) arow = N - 1;                       // clamp loads; store is guarded
    const float* Ap = H + (size_t)arow * IN_DIM + kb;
    const float* Bp = Wl + (n0 + r) * IN_DIM + kb;     // B[k][n] = weight[n][k]

    v8f acc = {};
#pragma unroll
    for (int k = 0; k < IN_DIM; k += 4) {
        v2f a = *(const v2f*)(Ap + k);
        v2f b = *(const v2f*)(Bp + k);
        acc = __builtin_amdgcn_wmma_f32_16x16x4_f32(
            /*neg_a=*/false, a, /*neg_b=*/false, b,
            /*c_mod=*/(short)0, acc, /*reuse_a=*/false, /*reuse_b=*/false);
    }

    int n = n0 + r;
    float bv = Bl[n];
    int mbase = m0 + ((lane >> 4) << 3);               // +0 or +8
#pragma unroll
    for (int rr = 0; rr < 8; ++rr) {
        int mrow = mbase + rr;
        if (mrow < N) out[(size_t)mrow * OUT_DIM + n] = acc[rr] + bv;
    }
}

extern "C" void kernel_launch(void* const* d_in, const int* in_sizes, int n_in,
                              void* d_out, int out_size, void* d_ws, size_t ws_size,
                              hipStream_t stream) {
    const int*   src    = (const int*)d_in[0];
    const int*   dst    = (const int*)d_in[1];
    const float* feat   = (const float*)d_in[2];
    const float* weight = (const float*)d_in[3];
    const float* bias   = (const float*)d_in[4];
    float*       outp   = (float*)d_out;

    const int E = in_sizes[0];
    const int N = in_sizes[2] / IN_DIM;

    // workspace layout: [norm: N (padded)] [h1: N*128] [h2: N*128]
    float* ws = (float*)d_ws;
    size_t npad = ((size_t)N + 1023) & ~(size_t)1023;
    float* norm = ws;
    float* h1 = ws + npad;
    float* h2 = h1 + (size_t)N * IN_DIM;

    const int ZB = 2048;          // zero-fill grid
    long normN4 = (long)npad / 4;
    long hN4    = ((long)N * IN_DIM) / 4;

    // 1) deg = 0; deg[dst[e]] += 1; norm = rsqrt(deg) (in place)
    sgc_zero_f4<<<ZB, 256, 0, stream>>>(norm, normN4);
    sgc_degree<<<(E + 255) / 256, 256, 0, stream>>>(dst, norm, E);
    sgc_norm<<<(N + 255) / 256, 256, 0, stream>>>(norm, N);

    int scatterBlocks = (int)(((long)E * 32 + 255) / 256);
    int scaleBlocks   = (int)(((long)N * 32 + 255) / 256);

    // 2) hop 1: h1 = norm * scatter(feat * norm[src])
    sgc_zero_f4<<<ZB, 256, 0, stream>>>(h1, hN4);
    sgc_scatter<<<scatterBlocks, 256, 0, stream>>>(src, dst, feat, norm, h1, E);
    sgc_scale<<<scaleBlocks, 256, 0, stream>>>(h1, norm, N);

    // 3) hop 2: h2 = norm * scatter(h1 * norm[src])
    sgc_zero_f4<<<ZB, 256, 0, stream>>>(h2, hN4);
    sgc_scatter<<<scatterBlocks, 256, 0, stream>>>(src, dst, h1, norm, h2, E);
    sgc_scale<<<scaleBlocks, 256, 0, stream>>>(h2, norm, N);

    // 4) out = h2 @ W^T + bias via fp32 WMMA
    int mtiles = (N + 15) / 16;
    int ntilesTotal = mtiles * (OUT_DIM / 16);
    int linBlocks = (ntilesTotal + 7) / 8;             // 8 waves per block
    sgc_linear_wmma<<<linBlocks, 256, 0, stream>>>(h2, weight, bias, outp, N);
}